// GCN_54262616818360
// MI455X (gfx1250) — compile-verified
//
#include <hip/hip_runtime.h>
#include <stdint.h>

typedef __attribute__((ext_vector_type(2))) float v2f;
typedef __attribute__((ext_vector_type(8))) float v8f;
typedef int v4i_vs __attribute__((vector_size(4 * sizeof(int))));

#define GCN_IN_F 512
#define GCN_HID  64
#define GCN_NCLS 32

#if defined(__has_builtin)
#  if __has_builtin(__builtin_amdgcn_global_load_async_to_lds_b128)
#    define HAVE_ASYNC_B128 1
#  endif
#  if __has_builtin(__builtin_amdgcn_s_wait_asynccnt)
#    define HAVE_WAIT_ASYNC 1
#  endif
#endif

#if defined(HAVE_ASYNC_B128)
// 16B global -> LDS async copy (tracked by ASYNCcnt).
__device__ __forceinline__ void async_copy_b128(const float* gp, float* lp) {
  __builtin_amdgcn_global_load_async_to_lds_b128(
      (__attribute__((address_space(1))) v4i_vs*)(uintptr_t)gp,   // AS1: same bits as flat
      (__attribute__((address_space(3))) v4i_vs*)lp,              // proper addrspacecast
      0, 0);
}
#endif

__device__ __forceinline__ void async_wait0() {
#if defined(HAVE_WAIT_ASYNC)
  __builtin_amdgcn_s_wait_asynccnt(0);
#elif defined(HAVE_ASYNC_B128)
  asm volatile("s_wait_asynccnt 0" ::: "memory");
#endif
}

// ---------------- degree / norm precompute ----------------

__global__ void k_deg_init(int* __restrict__ deg, int n) {
  int i = blockIdx.x * 256 + threadIdx.x;
  if (i < n) deg[i] = 1;  // self-loop
}

__global__ void k_deg_count(const int* __restrict__ dst, int* __restrict__ deg, int e) {
  int i = blockIdx.x * 256 + threadIdx.x;
  if (i < e) atomicAdd(&deg[dst[i]], 1);
}

__global__ void k_dinv(const int* __restrict__ deg, float* __restrict__ dinv, int n) {
  int i = blockIdx.x * 256 + threadIdx.x;
  if (i < n) dinv[i] = rsqrtf((float)deg[i]);  // deg >= 1 always
}

__global__ void k_enorm(const int* __restrict__ src, const int* __restrict__ dst,
                        const float* __restrict__ dinv, float* __restrict__ enorm, int e) {
  int i = blockIdx.x * 256 + threadIdx.x;
  if (i < e) enorm[i] = dinv[src[i]] * dinv[dst[i]];
}

// ---------------- WMMA f32 GEMM: Y[N x NOUT] = X[N x K] * W[K x NOUT] ----------------
// Block tile: BM=32 rows x NOUT cols. blockDim.x == (NOUT/16)*32; wave w owns N-tile w
// and computes BOTH 16-row M sub-tiles (shared B fragment -> 2 WMMAs per B load,
// two independent accumulation chains). A staged in LDS, +4 float pad (stride%64==4
// => conflict-free: bank = 4*lane + c).

template<int K, int NOUT>
__global__ void gcn_gemm(const float* __restrict__ X, const float* __restrict__ W,
                         float* __restrict__ Y, int N) {
  constexpr int BM  = 32;
  constexpr int LDA = K + 4;
  constexpr int NTHREADS = (NOUT / 16) * 32;
  __shared__ __attribute__((aligned(16))) float As[BM * LDA];

  const int m0  = blockIdx.x * BM;
  const int tid = threadIdx.x;

  // Stage the BM x K A-tile into LDS (128-bit granules).
  for (int i = tid; i < BM * (K / 4); i += NTHREADS) {
    int r = i / (K / 4);
    int c = (i % (K / 4)) * 4;
    int m = m0 + r; if (m >= N) m = N - 1;          // tail clamp (dup rows; stores masked)
    const float* gp = X + (size_t)m * K + c;
    float*       lp = &As[r * LDA + c];
#if defined(HAVE_ASYNC_B128)
    async_copy_b128(gp, lp);
#else
    *(float4*)lp = *(const float4*)gp;
#endif
  }
#if defined(HAVE_ASYNC_B128)
  async_wait0();
#endif
  __syncthreads();

  const int lane = tid & 31;
  const int n0   = (tid >> 5) * 16;                  // this wave's N tile
  const int row  = lane & 15;
  const int kh   = (lane >> 4) << 1;                 // 0 for lanes 0-15, 2 for 16-31

  v8f acc0 = {};                                     // rows m0 .. m0+15
  v8f acc1 = {};                                     // rows m0+16 .. m0+31
  const float* a0p = &As[row * LDA];
  const float* a1p = &As[(row + 16) * LDA];
  #pragma unroll 4
  for (int k0 = 0; k0 < K; k0 += 4) {
    // B 4x16 f32 fragment (shared by both M sub-tiles)
    v2f b;
    b[0] = W[(size_t)(k0 + kh) * NOUT + n0 + row];
    b[1] = W[(size_t)(k0 + kh + 1) * NOUT + n0 + row];
    // A 16x4 f32 fragments
    v2f a0; a0[0] = a0p[k0 + kh]; a0[1] = a0p[k0 + kh + 1];
    v2f a1; a1[0] = a1p[k0 + kh]; a1[1] = a1p[k0 + kh + 1];
    acc0 = __builtin_amdgcn_wmma_f32_16x16x4_f32(false, a0, false, b, (short)0, acc0, false, false);
    acc1 = __builtin_amdgcn_wmma_f32_16x16x4_f32(false, a1, false, b, (short)0, acc1, false, false);
  }

  // D layout: VGPR r -> M = r + 8*(lane>=16), N = lane&15
  const int hi = lane >> 4;
  const int nD = n0 + row;
  float* y0 = Y + (size_t)(m0 + hi * 8) * NOUT + nD;
  float* y1 = Y + (size_t)(m0 + 16 + hi * 8) * NOUT + nD;
  if (m0 + BM <= N) {                                // uniform full-tile fast path
    #pragma unroll
    for (int r = 0; r < 8; ++r) {
      y0[(size_t)r * NOUT] = acc0[r];
      y1[(size_t)r * NOUT] = acc1[r];
    }
  } else {
    #pragma unroll
    for (int r = 0; r < 8; ++r) {
      if (m0 + hi * 8 + r      < N) y0[(size_t)r * NOUT] = acc0[r];
      if (m0 + 16 + hi * 8 + r < N) y1[(size_t)r * NOUT] = acc1[r];
    }
  }
}

// ---------------- aggregation ----------------

// out[i,:] = xw[i,:] * dinv[i]^2   (self-loop term; also fully initializes d_out region)
template<int F>
__global__ void k_self_init(const float* __restrict__ xw, const float* __restrict__ dinv,
                            float* __restrict__ out, int n) {
  int idx = blockIdx.x * 256 + threadIdx.x;
  if (idx < n * F) {
    float d = dinv[idx / F];
    out[idx] = xw[idx] * d * d;
  }
}

// One wave per edge: lanes cover F features, hardware f32 atomics into L2-resident acc.
template<int F>
__global__ void k_scatter(const int* __restrict__ src, const int* __restrict__ dst,
                          const float* __restrict__ enorm, const float* __restrict__ xw,
                          float* __restrict__ out, int e) {
  int gw   = (int)((blockIdx.x * (unsigned)blockDim.x + threadIdx.x) >> 5);
  int lane = threadIdx.x & 31;
  if (gw >= e) return;
  int   s   = src[gw];
  int   d   = dst[gw];
  float nrm = enorm[gw];
  const float* srow = xw  + (size_t)s * F;
  float*       drow = out + (size_t)d * F;
  #pragma unroll
  for (int f = lane; f < F; f += 32)
    unsafeAtomicAdd(&drow[f], srow[f] * nrm);   // global_atomic_add_f32
}

template<int F, bool RELU>
__global__ void k_bias(float* __restrict__ out, const float* __restrict__ b, int n) {
  int idx = blockIdx.x * 256 + threadIdx.x;
  if (idx < n * F) {
    float v = out[idx] + b[idx & (F - 1)];
    out[idx] = RELU ? fmaxf(v, 0.0f) : v;
  }
}

// ---------------- launch ----------------

extern "C" void kernel_launch(void* const* d_in, const int* in_sizes, int n_in,
                              void* d_out, int out_size, void* d_ws, size_t ws_size,
                              hipStream_t stream) {
  const float* x  = (const float*)d_in[0];
  const int*   ei = (const int*)d_in[1];    // [2, E] row-major
  const float* W1 = (const float*)d_in[2];
  const float* b1 = (const float*)d_in[3];
  const float* W2 = (const float*)d_in[4];
  const float* b2 = (const float*)d_in[5];

  const int N = in_sizes[0] / GCN_IN_F;
  const int E = in_sizes[1] / 2;
  const int* src = ei;
  const int* dst = ei + E;

  float* h_out = (float*)d_out;                   // [N, 64]
  float* z_out = h_out + (size_t)N * GCN_HID;     // [N, 32]

  // workspace carve-out
  char* ws = (char*)d_ws;
  auto al = [](size_t v) { return (v + 255) & ~(size_t)255; };
  size_t o = 0;
  float* xw1   = (float*)(ws + o); o += al((size_t)N * GCN_HID  * 4);
  float* hw2   = (float*)(ws + o); o += al((size_t)N * GCN_NCLS * 4);
  float* dinv  = (float*)(ws + o); o += al((size_t)N * 4);
  int*   deg   = (int*)  (ws + o); o += al((size_t)N * 4);
  float* enorm = (float*)(ws + o);

  const int gN = (N + 255) / 256;
  const int gE = (E + 255) / 256;
  k_deg_init <<<gN, 256, 0, stream>>>(deg, N);
  k_deg_count<<<gE, 256, 0, stream>>>(dst, deg, E);
  k_dinv     <<<gN, 256, 0, stream>>>(deg, dinv, N);
  k_enorm    <<<gE, 256, 0, stream>>>(src, dst, dinv, enorm, E);

  const int gM  = (N + 31) / 32;                  // BM = 32
  const int gSc = (E + 7) / 8;                    // 8 waves / block, 1 wave / edge
  const int gF1 = (N * GCN_HID  + 255) / 256;
  const int gF2 = (N * GCN_NCLS + 255) / 256;

  // Layer 1: h = relu(agg(x @ W1) + b1)
  gcn_gemm<GCN_IN_F, GCN_HID><<<gM, 128, 0, stream>>>(x, W1, xw1, N);
  k_self_init<GCN_HID>        <<<gF1, 256, 0, stream>>>(xw1, dinv, h_out, N);
  k_scatter<GCN_HID>          <<<gSc, 256, 0, stream>>>(src, dst, enorm, xw1, h_out, E);
  k_bias<GCN_HID, true>       <<<gF1, 256, 0, stream>>>(h_out, b1, N);

  // Layer 2: z = agg(h @ W2) + b2
  gcn_gemm<GCN_HID, GCN_NCLS><<<gM, 64, 0, stream>>>(h_out, W2, hw2, N);
  k_self_init<GCN_NCLS>       <<<gF2, 256, 0, stream>>>(hw2, dinv, z_out, N);
  k_scatter<GCN_NCLS>         <<<gSc, 256, 0, stream>>>(src, dst, enorm, hw2, z_out, E);
  k_bias<GCN_NCLS, false>     <<<gF2, 256, 0, stream>>>(z_out, b2, N);
}